// SampleLayer_65386582115013
// MI455X (gfx1250) — compile-verified
//
#include <hip/hip_runtime.h>
#include <cstdint>

// Problem constants (match reference)
constexpr int kB   = 128;   // batch
constexpr int kS   = 256;   // samples per batch
constexpr int kImg = 224;   // image H=W
constexpr int kP   = 16;    // patch size
// patch = 3*16*16 = 768 dwords = 3072 bytes, contiguous in output at (b*kS+s)*3072

// ---------------------------------------------------------------------------
// Kernel 1: coordinates + independent bitonic sort of x and y (per batch).
// One block per batch, 256 threads (8 waves of 32).
// ---------------------------------------------------------------------------
__global__ __launch_bounds__(256) void coords_sort_kernel(
    const float* __restrict__ mu, const float* __restrict__ beta,
    const float* __restrict__ noise, int* __restrict__ sx, int* __restrict__ sy) {
  __shared__ int ax[kS];
  __shared__ int ay[kS];
  const int b = blockIdx.x;
  const int t = threadIdx.x;

  int zx, zy;
  {
    // JAX/XLA does NOT fuse mul+add here; forbid FMA contraction so the
    // rounded patch indices match the reference bit-exactly.
#pragma clang fp contract(off)
    const float nx = noise[(size_t)(b * kS + t) * 2 + 0];
    const float ny = noise[(size_t)(b * kS + t) * 2 + 1];
    float fx = beta[b * 2 + 0] * nx + mu[b * 2 + 0];
    float fy = beta[b * 2 + 1] * ny + mu[b * 2 + 1];
    fx = fminf(fmaxf(fx, 0.0f), 1.0f);
    fy = fminf(fmaxf(fy, 0.0f), 1.0f);
    // jnp.round == round-half-to-even == rintf in default RTE mode.
    float rx = fminf(fmaxf(rintf(fx * 224.0f), 0.0f), (float)(kImg - kP));
    float ry = fminf(fmaxf(rintf(fy * 224.0f), 0.0f), (float)(kImg - kP));
    zx = (int)rx;
    zy = (int)ry;
  }
  ax[t] = zx;
  ay[t] = zy;
  __syncthreads();

  // Bitonic sort, 256 elements, ascending; x and y sorted independently.
  for (int k = 2; k <= kS; k <<= 1) {
    for (int j = k >> 1; j > 0; j >>= 1) {
      const int ixj = t ^ j;
      if (ixj > t) {
        const bool up = ((t & k) == 0);
        int a0 = ax[t], a1 = ax[ixj];
        if ((a0 > a1) == up) { ax[t] = a1; ax[ixj] = a0; }
        int c0 = ay[t], c1 = ay[ixj];
        if ((c0 > c1) == up) { ay[t] = c1; ay[ixj] = c0; }
      }
      __syncthreads();
    }
  }
  sx[b * kS + t] = ax[t];
  sy[b * kS + t] = ay[t];
}

// ---------------------------------------------------------------------------
// Kernel 2: patch gather via the CDNA5 async global<->LDS engine.
// One block per (sample, batch): 192 threads (6 waves).
//   - 768 dwords of the patch pulled global->LDS with async b32 loads
//     (rows are 64B, only 4B-aligned in the source image).
//   - s_wait_asynccnt 0 + block barrier.
//   - 192 async b128 stores LDS->global: one contiguous, 16B-aligned 3KB
//     block per sample in d_out. Data never passes through VGPRs.
// ---------------------------------------------------------------------------
__global__ __launch_bounds__(192) void gather_patches_kernel(
    const float* __restrict__ img, const int* __restrict__ sxg,
    const int* __restrict__ syg, float* __restrict__ out) {
  __shared__ float patch[768];
  const int s = blockIdx.x;
  const int b = blockIdx.y;
  const int t = threadIdx.x;  // 0..191, all lanes active (EXEC all ones)

  const int x = sxg[b * kS + s];
  const int y = syg[b * kS + s];

  // Low 32 bits of the generic pointer == workgroup-relative LDS byte offset.
  const unsigned lds_base = (unsigned)(uintptr_t)(&patch[0]);

#pragma unroll
  for (int k = 0; k < 4; ++k) {
    const int d   = k * 192 + t;   // dword index within patch, 0..767
    const int c   = d >> 8;        // channel 0..2
    const int rem = d & 255;
    const int r   = rem >> 4;      // row in patch 0..15
    const int col = rem & 15;      // col in patch 0..15
    const unsigned goff =
        (unsigned)((((b * 3 + c) * kImg + (x + r)) * kImg + (y + col)) * 4);
    const unsigned loff = lds_base + (unsigned)(d * 4);
    asm volatile("global_load_async_to_lds_b32 %0, %1, %2"
                 :
                 : "v"(loff), "v"(goff), "s"(img)
                 : "memory");
  }

  // Wait for this wave's async loads, then make LDS visible across waves.
  asm volatile("s_wait_asynccnt 0" ::: "memory");
  __syncthreads();

  // Contiguous 3KB output block for this (b, s); 16B-aligned b128 stores.
  const unsigned ooff  = (unsigned)((b * kS + s) * 3072 + t * 16);
  const unsigned loff2 = lds_base + (unsigned)(t * 16);
  asm volatile("global_store_async_from_lds_b128 %0, %1, %2"
               :
               : "v"(ooff), "v"(loff2), "s"(out)
               : "memory");
  // S_ENDPGM performs an implicit wait-idle, covering the async stores.
}

// ---------------------------------------------------------------------------
extern "C" void kernel_launch(void* const* d_in, const int* in_sizes, int n_in,
                              void* d_out, int out_size, void* d_ws, size_t ws_size,
                              hipStream_t stream) {
  (void)in_sizes; (void)n_in; (void)out_size; (void)ws_size;
  const float* image = (const float*)d_in[0];
  const float* mu    = (const float*)d_in[1];
  const float* beta  = (const float*)d_in[2];
  const float* noise = (const float*)d_in[3];

  int* sx = (int*)d_ws;            // 128*256 ints
  int* sy = sx + kB * kS;          // 128*256 ints (total 256 KB scratch)

  coords_sort_kernel<<<kB, kS, 0, stream>>>(mu, beta, noise, sx, sy);
  gather_patches_kernel<<<dim3(kS, kB), 192, 0, stream>>>(
      image, sx, sy, (float*)d_out);
}